// RipEncoding_67551245631829
// MI455X (gfx1250) — compile-verified
//
#include <hip/hip_runtime.h>
#include <math.h>

// ---------------------------------------------------------------------------
// RipEncoding for MI455X (gfx1250, wave32)
//
// Stage 1/2: build 8x8 rip-map atlas (NV,1020,1020,16) f32 in d_ws (~666MB).
// Stage 3  : per-point projection uv = means*P^T and sigma^2 quadforms via
//            V_WMMA_F32_16X16X4_F32 (f32 WMMA keeps texel-coordinate precision;
//            f16 WMMA would lose ~0.25 texel). Writes (N,NV) float4 records
//            [u, v, lvl_u, lvl_v] into d_ws after the atlas.
// Stage 4  : quadlinear gather, 1 thread per (n,view), 16 taps x 4 float4
//            (global_load_b128) each -> HBM-bound, ~2.7GB of 64B gathers.
// ---------------------------------------------------------------------------

#define NVW   10
#define RESO  512
#define FDIMC 16
#define TATL  1020      // atlas side = sum(512>>l, l=0..7)
#define NPTS  262144

// OFFS(l) = 1024 - (1024>>l), SIZES(l) = 512>>l  (computed inline, no tables)
__device__ __forceinline__ int OFFS(int l)  { return 1024 - (1024 >> l); }
__device__ __forceinline__ int SIZEL(int l) { return RESO >> l; }

// Icosahedral projection bases P[v][k][d], k=0:p0, k=1:p1 (hand-evaluated from
// the reference formula: p0=normalize(-ay,ax,0), p1=a x p0hat).
static __device__ const float c_P[NVW][2][3] = {
  {{-0.70710678f, 0.70710678f, 0.f}, {-0.40824829f,-0.40824829f, 0.81649658f}},
  {{-0.70710678f, 0.70710678f, 0.f}, { 0.40824829f, 0.40824829f, 0.81649658f}},
  {{ 0.70710678f, 0.70710678f, 0.f}, {-0.40824829f, 0.40824829f, 0.81649658f}},
  {{ 0.70710678f, 0.70710678f, 0.f}, { 0.40824829f,-0.40824829f, 0.81649658f}},
  {{-1.f, 0.f, 0.f},                 { 0.f,-0.35682209f, 0.93417236f}},
  {{-1.f, 0.f, 0.f},                 { 0.f, 0.35682209f, 0.93417236f}},
  {{ 0.f, 1.f, 0.f},                 {-0.93417236f, 0.f, 0.35682209f}},
  {{ 0.f, 1.f, 0.f},                 { 0.93417236f, 0.f, 0.35682209f}},
  {{-0.35682209f, 0.93417236f, 0.f}, { 0.f, 0.f, 1.f}},
  {{ 0.35682209f, 0.93417236f, 0.f}, { 0.f, 0.f, 1.f}},
};

typedef __attribute__((ext_vector_type(2))) float v2f;
typedef __attribute__((ext_vector_type(8))) float v8f;

// ---------------------------------------------------------------------------
// Pass 1: width pooling. Fills atlas rows [0,512) (the l2=0 strip).
// atlas[v][y][OFFS(l1)+x] = mean_{k<2^l1} fm[v][y][(x<<l1)+k]
// One thread per (texel, float4-channel-group). fm (168MB) is L2-resident.
// ---------------------------------------------------------------------------
__global__ __launch_bounds__(256) void k_pool_w(const float* __restrict__ fm,
                                                float* __restrict__ atlas) {
  long long tid = (long long)blockIdx.x * 256 + threadIdx.x;
  int f4 = (int)(tid & 3);
  long long r = tid >> 2;
  int col = (int)(r % TATL); r /= TATL;
  int y   = (int)(r % RESO);
  int v   = (int)(r / RESO);
  if (v >= NVW) return;

  int l1 = 0, x = col;
  #pragma unroll
  for (int l = 0; l < 8; ++l)
    if (col >= OFFS(l)) { l1 = l; x = col - OFFS(l); }

  int cnt = 1 << l1;
  const float4* src = (const float4*)fm
      + (((long long)v * RESO + y) * RESO + ((long long)x << l1)) * 4 + f4;
  float4 acc = make_float4(0.f, 0.f, 0.f, 0.f);
  for (int k = 0; k < cnt; ++k) {
    float4 t = src[(long long)k * 4];            // next texel = +16 floats
    acc.x += t.x; acc.y += t.y; acc.z += t.z; acc.w += t.w;
  }
  float s = 1.0f / (float)cnt;
  acc.x *= s; acc.y *= s; acc.z *= s; acc.w *= s;
  ((float4*)atlas)[(((long long)v * TATL + y) * TATL + col) * 4 + f4] = acc;
}

// ---------------------------------------------------------------------------
// Pass 2: height pooling from the l2=0 strip (rows <512) into rows >=512.
// Reads and writes are disjoint row ranges -> safe in one kernel.
// ---------------------------------------------------------------------------
__global__ __launch_bounds__(256) void k_pool_h(float* __restrict__ atlas) {
  long long tid = (long long)blockIdx.x * 256 + threadIdx.x;
  const long long total = (long long)NVW * 508 * TATL * 4;
  if (tid >= total) return;
  int f4 = (int)(tid & 3);
  long long r = tid >> 2;
  int col = (int)(r % TATL); r /= TATL;
  int yr  = (int)(r % 508);
  int v   = (int)(r / 508);

  int l2 = 1, y = yr;
  #pragma unroll
  for (int l = 1; l < 8; ++l) {
    int base = OFFS(l) - RESO;                   // 0,256,384,448,480,496,504
    if (yr >= base) { l2 = l; y = yr - base; }
  }

  int cnt = 1 << l2;
  const float4* src = (const float4*)atlas
      + (((long long)v * TATL + ((long long)y << l2)) * TATL + col) * 4 + f4;
  float4 acc = make_float4(0.f, 0.f, 0.f, 0.f);
  for (int k = 0; k < cnt; ++k) {
    float4 t = src[(long long)k * TATL * 4];     // next row = +TATL texels
    acc.x += t.x; acc.y += t.y; acc.z += t.z; acc.w += t.w;
  }
  float s = 1.0f / (float)cnt;
  acc.x *= s; acc.y *= s; acc.z *= s; acc.w *= s;
  ((float4*)atlas)[(((long long)v * TATL + (OFFS(l2) + y)) * TATL + col) * 4 + f4] = acc;
}

// ---------------------------------------------------------------------------
// Projection via V_WMMA_F32_16X16X4_F32 (wave32).
// Per wave: 16 points. D = A(16x4) x B(4x16) + C.
//   uv   : A = means (K=3, padded to 4), B = P columns (c = 2v+j), 2 col-blocks.
//   q    : A = covs flattened (K=9 -> 3 chained K=4 WMMAs), B = outer(P_c).
// Fragment layouts per ISA 7.12.2 (f32 A 16x4: lane half splits K{0,1}/{2,3};
// D: VGPR j = rows j / j+8). lvl = clamp(0.5*log2(q)+10, 0, 7).
// Output record per (n,view): float4 [u, v, lvl_u, lvl_v].
// ---------------------------------------------------------------------------
__global__ __launch_bounds__(256) void kp_project(const float* __restrict__ means,
                                                  const float* __restrict__ covs,
                                                  float* __restrict__ uvlvl) {
  const int lane = threadIdx.x & 31;
  const int wav  = threadIdx.x >> 5;
  const int h    = lane >> 4;          // K-half for A/B, row-half for D
  const int m    = lane & 15;          // A row / B,D column
  const int pb   = (blockIdx.x * 8 + wav) * 16;
  const int pA   = pb + m;

  // A fragment for uv: K = 2h+j, means padded with 0 at K=3
  v2f a_uv;
  a_uv.x = h ? means[pA * 3 + 2] : means[pA * 3 + 0];
  a_uv.y = h ? 0.0f              : means[pA * 3 + 1];

  // A fragments for quadform: covflat[K], K=0..8, padded to 12 (3 K=4 chunks)
  v2f aq[3];
  #pragma unroll
  for (int t = 0; t < 3; ++t) {
    int k0 = 4 * t + 2 * h;
    aq[t].x = (k0     < 9) ? covs[pA * 9 + k0]     : 0.0f;
    aq[t].y = (k0 + 1 < 9) ? covs[pA * 9 + k0 + 1] : 0.0f;
  }

  #pragma unroll
  for (int b = 0; b < 2; ++b) {
    const int  c     = b * 16 + m;        // output column in [0,20)
    const bool valid = (c < 2 * NVW);
    const int  vv    = c >> 1, jj = c & 1;

    float pv[3] = {0.f, 0.f, 0.f};
    if (valid) { pv[0] = c_P[vv][jj][0]; pv[1] = c_P[vv][jj][1]; pv[2] = c_P[vv][jj][2]; }

    v2f b_uv;
    b_uv.x = (2 * h     < 3) ? pv[2 * h]     : 0.0f;
    b_uv.y = (2 * h + 1 < 3) ? pv[2 * h + 1] : 0.0f;

    float pout[9];
    #pragma unroll
    for (int kk = 0; kk < 9; ++kk) pout[kk] = pv[kk / 3] * pv[kk % 3];
    v2f bq[3];
    #pragma unroll
    for (int t = 0; t < 3; ++t) {
      int k0 = 4 * t + 2 * h;
      bq[t].x = (k0     < 9) ? pout[k0]     : 0.0f;
      bq[t].y = (k0 + 1 < 9) ? pout[k0 + 1] : 0.0f;
    }

    v8f zero = {};
    // (neg_a, A, neg_b, B, c_mod, C, reuse_a, reuse_b)
    v8f du = __builtin_amdgcn_wmma_f32_16x16x4_f32(false, a_uv,  false, b_uv,  (short)0, zero, false, false);
    v8f dq = __builtin_amdgcn_wmma_f32_16x16x4_f32(false, aq[0], false, bq[0], (short)0, zero, false, false);
    dq     = __builtin_amdgcn_wmma_f32_16x16x4_f32(false, aq[1], false, bq[1], (short)0, dq,   false, false);
    dq     = __builtin_amdgcn_wmma_f32_16x16x4_f32(false, aq[2], false, bq[2], (short)0, dq,   false, false);

    if (valid) {
      #pragma unroll
      for (int j = 0; j < 8; ++j) {
        int   p   = pb + h * 8 + j;                   // D row -> point
        float q   = dq[j];
        float lvl = 0.5f * log2f(q) + 10.0f;          // log2(sqrt(q)*2)+9
        lvl = fminf(fmaxf(lvl, 0.0f), 7.0f);
        uvlvl[p * 40 + vv * 4 + jj]     = du[j];
        uvlvl[p * 40 + vv * 4 + 2 + jj] = lvl;
      }
    }
  }
}

// ---------------------------------------------------------------------------
// Quadlinear gather: one thread per (n,view). 16 taps, 4 x global_load_b128
// per tap (a tap is exactly one 64B cacheline segment).
// ---------------------------------------------------------------------------
__device__ __forceinline__ void fma4(float4& a, const float4 b, const float w) {
  a.x += b.x * w; a.y += b.y * w; a.z += b.z * w; a.w += b.w * w;
}

__global__ __launch_bounds__(256) void k_gather(const float* __restrict__ uvlvl,
                                                const float* __restrict__ atlas,
                                                float* __restrict__ out) {
  const int t = blockIdx.x * 256 + threadIdx.x;   // flat (n,view), exact grid
  const int v = t % NVW;

  float4 r = ((const float4*)uvlvl)[t];
  const float u = r.x, w = r.y, lu = r.z, lv = r.w;

  float l0u = floorf(lu), l0v = floorf(lv);
  float fu = lu - l0u,    fv = lv - l0v;
  int iu0 = (int)l0u, iv0 = (int)l0v;
  int iu1 = min(iu0 + 1, 7), iv1 = min(iv0 + 1, 7);

  const int   lxs[4] = {iu0, iu1, iu0, iu1};
  const int   lys[4] = {iv0, iv0, iv1, iv1};
  const float wls[4] = {(1.f - fu) * (1.f - fv), fu * (1.f - fv),
                        (1.f - fu) * fv,         fu * fv};

  float4 acc[4];
  #pragma unroll
  for (int f = 0; f < 4; ++f) acc[f] = make_float4(0.f, 0.f, 0.f, 0.f);

  #pragma unroll
  for (int s = 0; s < 4; ++s) {
    const int lx = lxs[s], ly = lys[s];
    const float wgt = wls[s];
    const int W = SIZEL(lx), H = SIZEL(ly);
    const int ox = OFFS(lx), oy = OFFS(ly);

    float x = fminf(fmaxf((u + 1.0f) * 0.5f * (float)W - 0.5f, 0.0f), (float)(W - 1));
    float y = fminf(fmaxf((w + 1.0f) * 0.5f * (float)H - 0.5f, 0.0f), (float)(H - 1));
    float x0f = floorf(x), y0f = floorf(y);
    float wx = x - x0f,    wy = y - y0f;
    int x0 = (int)x0f, y0 = (int)y0f;
    int x1 = min(x0 + 1, W - 1), y1 = min(y0 + 1, H - 1);

    const long long rb = (long long)v * TATL;
    const float4* a00 = (const float4*)atlas + ((rb + oy + y0) * TATL + ox + x0) * 4;
    const float4* a01 = (const float4*)atlas + ((rb + oy + y0) * TATL + ox + x1) * 4;
    const float4* a10 = (const float4*)atlas + ((rb + oy + y1) * TATL + ox + x0) * 4;
    const float4* a11 = (const float4*)atlas + ((rb + oy + y1) * TATL + ox + x1) * 4;

    const float w00 = wgt * (1.f - wx) * (1.f - wy);
    const float w01 = wgt * wx * (1.f - wy);
    const float w10 = wgt * (1.f - wx) * wy;
    const float w11 = wgt * wx * wy;

    #pragma unroll
    for (int f = 0; f < 4; ++f) {
      fma4(acc[f], a00[f], w00);
      fma4(acc[f], a01[f], w01);
      fma4(acc[f], a10[f], w10);
      fma4(acc[f], a11[f], w11);
    }
  }

  float4* o = (float4*)out + (long long)t * 4;    // out[n][v*16 + f]
  #pragma unroll
  for (int f = 0; f < 4; ++f) o[f] = acc[f];
}

// ---------------------------------------------------------------------------
extern "C" void kernel_launch(void* const* d_in, const int* in_sizes, int n_in,
                              void* d_out, int out_size, void* d_ws, size_t ws_size,
                              hipStream_t stream) {
  const float* means = (const float*)d_in[0];   // (N,3)
  const float* covs  = (const float*)d_in[1];   // (N,3,3)
  const float* fm    = (const float*)d_in[2];   // (NV,512,512,16)

  const size_t atlasBytes = (size_t)NVW * TATL * TATL * FDIMC * sizeof(float); // 665,856,000 (16B aligned)
  float* atlas = (float*)d_ws;
  float* uvlvl = (float*)((char*)d_ws + atlasBytes);                            // (N,NV,4) f32
  float* out   = (float*)d_out;

  // Stage 1: width pooling -> l2=0 strip.  10*512*1020*4 / 256 = 81600 blocks.
  k_pool_w<<<81600, 256, 0, stream>>>(fm, atlas);
  // Stage 2: height pooling -> remaining strips.
  k_pool_h<<<80963, 256, 0, stream>>>(atlas);
  // Stage 3: WMMA projection (independent of atlas build).
  kp_project<<<NPTS / 128, 256, 0, stream>>>(means, covs, uvlvl);
  // Stage 4: gather.  N*NV/256 = 10240 blocks, exact.
  k_gather<<<(NPTS * NVW) / 256, 256, 0, stream>>>(uvlvl, atlas, out);
}